// XiLoss_90005334655187
// MI455X (gfx1250) — compile-verified
//
#include <hip/hip_runtime.h>
#include <hip/hip_bf16.h>

#define NN 4096
#define INV_TAU 100.0f
// -(1/tau) * log2(e): fold the softmax temperature and the exp->exp2 conversion
// into one constant so the hot loop is sub, square, one mul, v_exp_f32.
#define NEG_INVTAU_LOG2E (-144.26950408889634f)

typedef __attribute__((ext_vector_type(2))) float v2f;
typedef __attribute__((ext_vector_type(8))) float v8f;

// ---------------- Kernel 1: bitonic sort of preds (ascending) into xs ----------------
__global__ void xi_sort_preds(const float* __restrict__ preds, float* __restrict__ xs) {
  __shared__ float s[NN];
  const int t = threadIdx.x; // 1024 threads
  for (int i = t; i < NN; i += 1024) s[i] = preds[i];
  __syncthreads();
  for (int k = 2; k <= NN; k <<= 1) {
    for (int j = k >> 1; j > 0; j >>= 1) {
      for (int i = t; i < NN; i += 1024) {
        int ixj = i ^ j;
        if (ixj > i) {
          bool up = ((i & k) == 0);
          float a = s[i], b = s[ixj];
          if ((a > b) == up) { s[i] = b; s[ixj] = a; }
        }
      }
      __syncthreads();
    }
  }
  for (int i = t; i < NN; i += 1024) xs[i] = s[i];
}

// ---- Kernel 2: y[i] = softmax_j(-(xs_i - preds_j)^2/tau) . target  via WMMA f32 16x16x4
// Each wave owns 16 output rows. Per K-chunk of 4 columns j:
//   A (16x4): A[m][k] = exp2((xs[i0+m]-preds[j0+k])^2 * -100*log2e)  (lane L: m=L%16, k=2*(L/16)+v)
//   B (4x16): col0 = target[j0+k], col1 = 1, rest 0                   (lane L: n=L%16, k=2*(L/16)+v)
//   C (16x16 f32): col0 accumulates numerator, col1 denominator.
// B is built branch-free (fma(bmask, t, bone)) so the hot loop has no exec-mask churn.
__global__ void xi_softperm_matvec(const float* __restrict__ preds,
                                   const float* __restrict__ target,
                                   const float* __restrict__ xs,
                                   float* __restrict__ y) {
  __shared__ float ldsX[NN];
  __shared__ float ldsT[NN];
  const int t = threadIdx.x; // 256
  for (int i = t; i < NN; i += 256) { ldsX[i] = preds[i]; ldsT[i] = target[i]; }
  __syncthreads();

  const int wave = t >> 5;
  const int lane = t & 31;
  const int m    = lane & 15;  // M for A, N for B (same striping)
  const int half = lane >> 4;
  const int i0   = (blockIdx.x * 8 + wave) * 16;

  const float xsm   = xs[i0 + m];
  const float bmask = (m == 0) ? 1.0f : 0.0f;  // column 0 of B carries target
  const float bone  = (m == 1) ? 1.0f : 0.0f;  // column 1 of B carries ones
  const int   kbase = half * 2;

  v8f c = {0.f, 0.f, 0.f, 0.f, 0.f, 0.f, 0.f, 0.f};

  for (int j0 = 0; j0 < NN; j0 += 16) {
#pragma unroll
    for (int u = 0; u < 4; ++u) {
      v2f a, b;
#pragma unroll
      for (int v = 0; v < 2; ++v) {
        const int idx = j0 + u * 4 + kbase + v;
        const float d = xsm - ldsX[idx];
        // row max logit is exactly 0 (xs_i is one of preds_j) -> no stabilization needed
        a[v] = __builtin_amdgcn_exp2f((d * d) * NEG_INVTAU_LOG2E);
        b[v] = __builtin_fmaf(bmask, ldsT[idx], bone);
      }
      c = __builtin_amdgcn_wmma_f32_16x16x4_f32(false, a, false, b, (short)0, c,
                                                false, false);
    }
  }

  // C/D layout: VGPR r -> lanes 0-15: (M=r, N=lane); lanes 16-31: (M=8+r, N=lane-16)
  const int src0 = (lane < 16) ? 0 : 16;
#pragma unroll
  for (int rdx = 0; rdx < 8; ++rdx) {
    float num = __shfl(c[rdx], src0, 32);
    float den = __shfl(c[rdx], src0 + 1, 32);
    if (lane == 0 || lane == 16) {
      int row = (lane == 0) ? rdx : (8 + rdx);
      y[i0 + row] = num / den;
    }
  }
}

// ---- Kernel 3: soft_rank(y, tau) + xi reduction (one workgroup) ----
// Sort key = -(100*y) ascending (== s descending), carry original index.
// PAV non-increasing isotonic fit of z[i] = s_sorted[i] - (N - i); block stack in d_ws.
// r[orig_idx] = s_sorted - blockmean; then xi = 1 - 3*sum|r[i+1]-r[i]|/(N^2-1).
__global__ void xi_softrank_reduce(const float* __restrict__ y,
                                   float* __restrict__ bsum, int* __restrict__ bcnt,
                                   float* __restrict__ out) {
  __shared__ float key[NN];
  __shared__ int   val[NN];
  __shared__ float rr[NN];
  __shared__ float red[1024];
  const int t = threadIdx.x; // 1024
  for (int i = t; i < NN; i += 1024) { key[i] = -(y[i] * INV_TAU); val[i] = i; }
  __syncthreads();

  for (int k = 2; k <= NN; k <<= 1) {
    for (int j = k >> 1; j > 0; j >>= 1) {
      for (int i = t; i < NN; i += 1024) {
        int ixj = i ^ j;
        if (ixj > i) {
          bool up = ((i & k) == 0);
          float a = key[i], b = key[ixj];
          if ((a > b) == up) {
            key[i] = b; key[ixj] = a;
            int va = val[i]; val[i] = val[ixj]; val[ixj] = va;
          }
        }
      }
      __syncthreads();
    }
  }

  if (t == 0) {
    // PAV: merge adjacent blocks while prev_mean < cur_mean (non-increasing fit)
    int nb = 0;
    for (int i = 0; i < NN; ++i) {
      float ssorted = -key[i];
      float z = ssorted - (float)(NN - i);
      float csum = z; int ccnt = 1;
      while (nb > 0 && bsum[nb - 1] * (float)ccnt < csum * (float)bcnt[nb - 1]) {
        csum += bsum[nb - 1]; ccnt += bcnt[nb - 1]; --nb;
      }
      bsum[nb] = csum; bcnt[nb] = ccnt; ++nb;
    }
    // expand + scatter ranks back to original order
    int pos = 0;
    for (int b = 0; b < nb; ++b) {
      float mean = bsum[b] / (float)bcnt[b];
      int cnt = bcnt[b];
      for (int q = 0; q < cnt; ++q) {
        float ssorted = -key[pos];
        rr[val[pos]] = ssorted - mean;
        ++pos;
      }
    }
  }
  __syncthreads();

  float acc = 0.f;
  for (int i = t; i < NN - 1; i += 1024) acc += fabsf(rr[i + 1] - rr[i]);
  red[t] = acc;
  __syncthreads();
  for (int sft = 512; sft > 0; sft >>= 1) {
    if (t < sft) red[t] += red[t + sft];
    __syncthreads();
  }
  if (t == 0) {
    float xi = 1.0f - 3.0f * red[0] / (float)(NN * NN - 1);
    out[0] = -xi;  // loss = -WEIGHT * xi, WEIGHT = 1
    out[1] = xi;
  }
}

extern "C" void kernel_launch(void* const* d_in, const int* in_sizes, int n_in,
                              void* d_out, int out_size, void* d_ws, size_t ws_size,
                              hipStream_t stream) {
  const float* preds  = (const float*)d_in[0];
  const float* target = (const float*)d_in[1];
  float* out = (float*)d_out;

  float* ws   = (float*)d_ws;
  float* xs   = ws;                  // [0,      4096) floats
  float* yv   = ws + NN;             // [4096,   8192) floats
  float* bsum = ws + 2 * NN;         // [8192,  12288) floats (PAV block sums)
  int*   bcnt = (int*)(ws + 3 * NN); // [12288, 16384) ints (PAV block counts)

  xi_sort_preds<<<1, 1024, 0, stream>>>(preds, xs);
  xi_softperm_matvec<<<32, 256, 0, stream>>>(preds, target, xs, yv);
  xi_softrank_reduce<<<1, 1024, 0, stream>>>(yv, bsum, bcnt, out);
}